// TMPHN_927712936182
// MI455X (gfx1250) — compile-verified
//
#include <hip/hip_runtime.h>
#include <hip/hip_bf16.h>
#include <math.h>

// Problem constants (match reference)
constexpr int N_NODES = 100000;
constexpr int D       = 256;
constexpr int B_BATCH = 20000;
constexpr int E_EDGES = 16;
constexpr int K_NODES = 3;
constexpr int H_DIM   = 256;
constexpr int C_CLS   = 40;

typedef __attribute__((ext_vector_type(2))) float        v2f;
typedef __attribute__((ext_vector_type(8))) float        v8f;
typedef __attribute__((ext_vector_type(4))) unsigned int v4u;
typedef __attribute__((ext_vector_type(4))) int          v4i;
typedef __attribute__((ext_vector_type(8))) int          v8i;

// ---------------------------------------------------------------------------
// Kernel 1: hypergraph message passing.
// Block = 256 threads = 4 groups of 64; each 64-thread group handles one node b.
// Thread tg owns float4 at feature offset tg*4 (so all gathers are B128).
// Edge e+1's three gathers are issued before edge e's reduction (SW pipeline).
// ---------------------------------------------------------------------------
__global__ __launch_bounds__(256) void tmphn_msgpass(
    const int* __restrict__ nodes,
    const int* __restrict__ structure,   // [B, E, K] int32
    const float* __restrict__ X,         // [N, D]
    const float* __restrict__ w_att_w,   // [1]
    const float* __restrict__ w_att_b,   // [1]
    float* __restrict__ hbuf)            // [B, D] output: self + neigh
{
    __shared__ int   sidx[4][E_EDGES * K_NODES];
    __shared__ int   snode[4];
    __shared__ float swsum[2][4][2];     // [parity][group][wave-in-group]

    const int g   = threadIdx.x >> 6;    // group 0..3
    const int tg  = threadIdx.x & 63;    // thread in group
    const int b   = blockIdx.x * 4 + g;
    const int wg  = tg >> 5;             // wave within group (0/1)
    const int ln  = tg & 31;

    if (tg < E_EDGES * K_NODES) sidx[g][tg] = structure[b * (E_EDGES * K_NODES) + tg];
    if (tg == 0) snode[g] = nodes[b];
    __syncthreads();

    const float waw = w_att_w[0];
    const float wab = w_att_b[0];

    float n0 = 0.f, n1 = 0.f, n2 = 0.f, n3 = 0.f;   // neigh accumulator
    const int doff = tg * 4;

    // prime the pipeline with edge 0's gathers
    float4 f0 = *reinterpret_cast<const float4*>(X + (long)sidx[g][0] * D + doff);
    float4 f1 = *reinterpret_cast<const float4*>(X + (long)sidx[g][1] * D + doff);
    float4 f2 = *reinterpret_cast<const float4*>(X + (long)sidx[g][2] * D + doff);

    for (int e = 0; e < E_EDGES; ++e) {
        float4 p0, p1, p2;
        if (e + 1 < E_EDGES) {           // issue next edge's gathers early
            p0 = *reinterpret_cast<const float4*>(X + (long)sidx[g][e * 3 + 3] * D + doff);
            p1 = *reinterpret_cast<const float4*>(X + (long)sidx[g][e * 3 + 4] * D + doff);
            p2 = *reinterpret_cast<const float4*>(X + (long)sidx[g][e * 3 + 5] * D + doff);
        }

        // sum over this thread's 4 features of sum_i (f_i - mean)^2
        float vs = 0.f;
        {
            const float m0 = (f0.x + f1.x + f2.x) * (1.f / 3.f);
            const float m1 = (f0.y + f1.y + f2.y) * (1.f / 3.f);
            const float m2 = (f0.z + f1.z + f2.z) * (1.f / 3.f);
            const float m3 = (f0.w + f1.w + f2.w) * (1.f / 3.f);
            float d;
            d = f0.x - m0; vs += d * d; d = f1.x - m0; vs += d * d; d = f2.x - m0; vs += d * d;
            d = f0.y - m1; vs += d * d; d = f1.y - m1; vs += d * d; d = f2.y - m1; vs += d * d;
            d = f0.z - m2; vs += d * d; d = f1.z - m2; vs += d * d; d = f2.z - m2; vs += d * d;
            d = f0.w - m3; vs += d * d; d = f1.w - m3; vs += d * d; d = f2.w - m3; vs += d * d;
        }
        // wave32 reduction, then combine the group's two waves through LDS
        #pragma unroll
        for (int off = 16; off > 0; off >>= 1) vs += __shfl_xor(vs, off, 32);
        const int par = e & 1;
        if (ln == 0) swsum[par][g][wg] = vs;
        __syncthreads();
        const float edge_var = (swsum[par][g][0] + swsum[par][g][1]) * (1.f / 768.f); // /(K*D)
        const float att = 1.f / (1.f + __expf(-(waw * edge_var + wab)));

        // msg = f0 * f1 (prod over first K-1 nodes); neigh += att * msg
        n0 = fmaf(att, f0.x * f1.x, n0);
        n1 = fmaf(att, f0.y * f1.y, n1);
        n2 = fmaf(att, f0.z * f1.z, n2);
        n3 = fmaf(att, f0.w * f1.w, n3);

        f0 = p0; f1 = p1; f2 = p2;
    }

    const float4 sf = *reinterpret_cast<const float4*>(X + (long)snode[g] * D + doff);
    float4 h;
    h.x = sf.x + n0; h.y = sf.y + n1; h.z = sf.z + n2; h.w = sf.w + n3;
    *reinterpret_cast<float4*>(hbuf + (long)b * D + doff) = h;
}

// ---------------------------------------------------------------------------
// Tensor Data Mover: DMA a [tile_d1 x tile_d0] fp32 tile (row-major, row
// stride `stride0` elements) from global into LDS at byte offset lds_off,
// inserting 1 dword of padding after every 256 dwords (pitch 256 -> 257).
// Descriptor layout per CDNA5 ISA ch.8 (groups 0/1; groups 2/3 zero for 2-D).
// Rows beyond tensor_d1 read as zero (OOB masking for the last cls tile).
// ---------------------------------------------------------------------------
__device__ __forceinline__ void tdm_load_tile_f32(
    const float* gptr, unsigned lds_off,
    unsigned tensor_d0, unsigned tensor_d1,
    unsigned tile_d0, unsigned tile_d1, unsigned stride0)
{
    const unsigned long long ga = (unsigned long long)(uintptr_t)gptr;
    v4u g0;
    g0[0] = 1u;                                              // count=1 (valid), user mode
    g0[1] = lds_off;                                         // lds_addr (bytes)
    g0[2] = (unsigned)(ga & 0xFFFFFFFFu);                    // global_addr[31:0]
    g0[3] = (unsigned)((ga >> 32) & 0x01FFFFFFu) | (2u << 30); // addr[56:32] | type=2
    v8i g1;
    g1[0] = (int)((2u << 16) | (1u << 20) | (7u << 22));     // data_size=4B, pad_en, pad every 256dw
    g1[1] = (int)((tensor_d0 & 0xFFFFu) << 16);              // [63:48] tensor_dim0 lo16
    g1[2] = (int)(((tensor_d0 >> 16) & 0xFFFFu) |            // [79:64] tensor_dim0 hi16
                  ((tensor_d1 & 0xFFFFu) << 16));            // [95:80] tensor_dim1 lo16
    g1[3] = (int)(((tensor_d1 >> 16) & 0xFFFFu) |            // [111:96] tensor_dim1 hi16
                  ((tile_d0 & 0xFFFFu) << 16));              // [127:112] tile_dim0
    g1[4] = (int)(tile_d1 & 0xFFFFu);                        // [143:128] tile_dim1 (tile_dim2=0)
    g1[5] = (int)stride0;                                    // tensor_dim0_stride[31:0]
    g1[6] = 0;                                               // stride0 hi16, stride1 lo16
    g1[7] = 0;
    const v4i gz = {0, 0, 0, 0};
#if __clang_major__ >= 23
    const v8i gz8 = {0, 0, 0, 0, 0, 0, 0, 0};
    __builtin_amdgcn_tensor_load_to_lds(g0, g1, gz, gz, gz8, 0);
#else
    __builtin_amdgcn_tensor_load_to_lds(g0, g1, gz, gz, 0);
#endif
}

// ---------------------------------------------------------------------------
// Kernel 2: fused GEMM head with V_WMMA_F32_16X16X4_F32 + TDM tile staging.
// Block = 128 threads (4 waves), one 16-row tile of B per block.
//   enc = (1-beta)*relu(h @ linW^T + linb) + beta*(self @ skipW^T)
//   out = log_softmax(enc @ clsW^T + clsb)
// Each 16x16 output tile: 4 waves split K=256 (64 each -> 16 WMMAs), partials
// reduced through LDS with the epilogue fused in. Weight/h tiles arrive via
// tensor_load_to_lds (wave 0 issues; TENSORcnt wait; barrier).
// ---------------------------------------------------------------------------
#define A_PITCH 257

__device__ __forceinline__ v8f wmma_tile_f32(const float* sAacc, const float* sWrm,
                                             int mrow, int half, int kbase)
{
    v8f acc = {0.f, 0.f, 0.f, 0.f, 0.f, 0.f, 0.f, 0.f};
    #pragma unroll
    for (int kk = 0; kk < 64; kk += 4) {
        const int k0 = kbase + kk + 2 * half;
        v2f a, bb;
        a.x  = sAacc[mrow * A_PITCH + k0];
        a.y  = sAacc[mrow * A_PITCH + k0 + 1];
        bb.x = sWrm[mrow * A_PITCH + k0];       // B[k][n] = W[j0+n][k], row-major tile
        bb.y = sWrm[mrow * A_PITCH + k0 + 1];
        acc = __builtin_amdgcn_wmma_f32_16x16x4_f32(false, a, false, bb,
                                                    (short)0, acc, false, false);
    }
    return acc;
}

__global__ __launch_bounds__(128) void tmphn_head(
    const float* __restrict__ hbuf,
    const float* __restrict__ X,
    const int*   __restrict__ nodes,
    const float* __restrict__ linW,  const float* __restrict__ linb,
    const float* __restrict__ skipW,
    const float* __restrict__ clsW,  const float* __restrict__ clsb,
    const float* __restrict__ beta_p,
    float* __restrict__ out)
{
    __shared__ float sA[16 * A_PITCH];       // A tile (h, then self feats)
    __shared__ float sE[16 * A_PITCH];       // enc accumulator (A tile for cls GEMM)
    __shared__ float sW[16 * A_PITCH];       // weight tile, row-major: sW[n*257 + k] = W[j0+n][k]
    __shared__ float sP[4][16][16];          // split-K partials
    __shared__ float sL[16][48];             // logits (C padded to 48)
    __shared__ float sLse[16];
    __shared__ int   snodes[16];

    const int tid  = threadIdx.x;
    const int wave = tid >> 5;
    const int lane = tid & 31;
    const int mrow = lane & 15;
    const int half = lane >> 4;
    const int b0   = blockIdx.x * 16;
    const float beta = beta_p[0];
    const int kbase = wave * 64;

    const unsigned sA_off = (unsigned)(uintptr_t)(void*)sA;  // LDS byte offsets
    const unsigned sW_off = (unsigned)(uintptr_t)(void*)sW;

    // ---- DMA the h tile (16 x 256, contiguous rows) into sA via TDM
    if (wave == 0) {
        tdm_load_tile_f32(hbuf + (long)b0 * D, sA_off, D, 16, D, 16, D);
        __builtin_amdgcn_s_wait_tensorcnt(0);
    }
    if (tid < 16) snodes[tid] = nodes[b0 + tid];
    __syncthreads();

    // ---- GEMM 1: sE = (1-beta) * relu(h @ linW^T + linb)
    for (int jt = 0; jt < H_DIM / 16; ++jt) {
        const int j0 = jt * 16;
        if (wave == 0) {
            tdm_load_tile_f32(linW + (long)j0 * D, sW_off, D, 16, D, 16, D);
            __builtin_amdgcn_s_wait_tensorcnt(0);
        }
        __syncthreads();
        const v8f acc = wmma_tile_f32(sA, sW, mrow, half, kbase);
        #pragma unroll
        for (int v = 0; v < 8; ++v) sP[wave][v + 8 * half][mrow] = acc[v];
        __syncthreads();
        for (int idx = tid; idx < 256; idx += 128) {
            const int m = idx >> 4, n = idx & 15;
            float s = sP[0][m][n] + sP[1][m][n] + sP[2][m][n] + sP[3][m][n];
            s += linb[j0 + n];
            s = s > 0.f ? s : 0.f;
            sE[m * A_PITCH + j0 + n] = (1.f - beta) * s;
        }
        __syncthreads();
    }

    // ---- reload sA with self features (gathered rows; X is L2-resident)
    for (int idx = tid; idx < 16 * D; idx += 128) {
        const int m = idx >> 8, k = idx & 255;
        sA[m * A_PITCH + k] = X[(long)snodes[m] * D + k];
    }
    __syncthreads();

    // ---- GEMM 2: sE += beta * (self @ skipW^T)
    for (int jt = 0; jt < H_DIM / 16; ++jt) {
        const int j0 = jt * 16;
        if (wave == 0) {
            tdm_load_tile_f32(skipW + (long)j0 * D, sW_off, D, 16, D, 16, D);
            __builtin_amdgcn_s_wait_tensorcnt(0);
        }
        __syncthreads();
        const v8f acc = wmma_tile_f32(sA, sW, mrow, half, kbase);
        #pragma unroll
        for (int v = 0; v < 8; ++v) sP[wave][v + 8 * half][mrow] = acc[v];
        __syncthreads();
        for (int idx = tid; idx < 256; idx += 128) {
            const int m = idx >> 4, n = idx & 15;
            const float s = sP[0][m][n] + sP[1][m][n] + sP[2][m][n] + sP[3][m][n];
            sE[m * A_PITCH + j0 + n] += beta * s;
        }
        __syncthreads();
    }

    // ---- GEMM 3: logits = enc @ clsW^T + clsb (3 col tiles; TDM OOB-zeros rows >= C)
    for (int ct = 0; ct < 3; ++ct) {
        const int j0 = ct * 16;
        if (wave == 0) {
            const unsigned rows_left = (unsigned)(C_CLS - j0);         // 40,24,8
            tdm_load_tile_f32(clsW + (long)j0 * H_DIM, sW_off,
                              H_DIM, rows_left, H_DIM, 16, H_DIM);
            __builtin_amdgcn_s_wait_tensorcnt(0);
        }
        __syncthreads();
        const v8f acc = wmma_tile_f32(sE, sW, mrow, half, kbase);
        #pragma unroll
        for (int v = 0; v < 8; ++v) sP[wave][v + 8 * half][mrow] = acc[v];
        __syncthreads();
        for (int idx = tid; idx < 256; idx += 128) {
            const int m = idx >> 4, n = idx & 15;
            if (j0 + n < C_CLS) {
                const float s = sP[0][m][n] + sP[1][m][n] + sP[2][m][n] + sP[3][m][n];
                sL[m][j0 + n] = s + clsb[j0 + n];
            }
        }
        __syncthreads();
    }

    // ---- log_softmax over C=40 (one thread per row; negligible work)
    if (tid < 16) {
        float mx = -INFINITY;
        for (int c = 0; c < C_CLS; ++c) mx = fmaxf(mx, sL[tid][c]);
        float se = 0.f;
        for (int c = 0; c < C_CLS; ++c) se += __expf(sL[tid][c] - mx);
        sLse[tid] = mx + __logf(se);
    }
    __syncthreads();
    for (int idx = tid; idx < 16 * C_CLS; idx += 128) {
        const int m = idx / C_CLS, c = idx % C_CLS;
        out[(long)(b0 + m) * C_CLS + c] = sL[m][c] - sLse[m];
    }
}

// ---------------------------------------------------------------------------
extern "C" void kernel_launch(void* const* d_in, const int* in_sizes, int n_in,
                              void* d_out, int out_size, void* d_ws, size_t ws_size,
                              hipStream_t stream) {
    const int*   nodes     = (const int*)  d_in[0];
    const int*   structure = (const int*)  d_in[1];
    const float* X         = (const float*)d_in[2];
    const float* beta      = (const float*)d_in[3];
    const float* w_att_w   = (const float*)d_in[4];
    const float* w_att_b   = (const float*)d_in[5];
    const float* linW      = (const float*)d_in[6];
    const float* linb      = (const float*)d_in[7];
    const float* skipW     = (const float*)d_in[8];
    const float* clsW      = (const float*)d_in[9];
    const float* clsb      = (const float*)d_in[10];
    float* out  = (float*)d_out;
    float* hbuf = (float*)d_ws;   // [B, D] fp32 = 20.48 MB scratch

    tmphn_msgpass<<<B_BATCH / 4, 256, 0, stream>>>(nodes, structure, X,
                                                   w_att_w, w_att_b, hbuf);
    tmphn_head<<<B_BATCH / 16, 128, 0, stream>>>(hbuf, X, nodes,
                                                 linW, linb, skipW, clsW, clsb,
                                                 beta, out);
}